// CNBiLSTMCRFNER_17686675325404
// MI455X (gfx1250) — compile-verified
//
#include <hip/hip_runtime.h>
#include <hip/hip_bf16.h>
#include <math.h>

typedef __attribute__((ext_vector_type(2))) float v2f;
typedef __attribute__((ext_vector_type(8))) float v8f;

__device__ __forceinline__ v8f wmma4(v2f a, v2f b, v8f c) {
  // D = A(16x4,f32) x B(4x16,f32) + C(16x16,f32)
  return __builtin_amdgcn_wmma_f32_16x16x4_f32(
      /*neg_a=*/false, a, /*neg_b=*/false, b,
      /*c_mod=*/(short)0, c, /*reuse_a=*/false, /*reuse_b=*/false);
}

#define NB   64      // batch
#define NL   256     // seq len
#define NE   100     // embed dim (per table)
#define NK   200     // 2E
#define NH   100     // hidden per direction
#define NG   400     // 4H
#define NC   33      // num tags

// ---------------- workspace layout (floats) ----------------
#define X_OFF      ((size_t)0)                     // [16384][200]
#define GF_OFF     (X_OFF + (size_t)16384*NK)      // [16384][400]
#define GB_OFF     (GF_OFF + (size_t)16384*NG)     // [16384][400]
#define HF_OFF     (GB_OFF + (size_t)16384*NG)     // [16384][100]
#define HB_OFF     (HF_OFF + (size_t)16384*NH)     // [16384][100]
#define LOGITS_OFF (HB_OFF + (size_t)16384*NH)     // [16384][33]
#define PERB_OFF   (LOGITS_OFF + (size_t)16384*NC) // [64]

// ---------------- 1: embedding gather ----------------
__global__ void __launch_bounds__(256)
embed_kernel(const int* __restrict__ chars, const int* __restrict__ bigrams,
             const float* __restrict__ char_tab, const float* __restrict__ bigram_tab,
             float* __restrict__ x) {
  int idx = blockIdx.x * 256 + threadIdx.x;      // 16384*200 threads
  int row = idx / NK, col = idx - row * NK;
  if (row >= NB * NL) return;
  float v;
  if (col < NE) v = char_tab[(size_t)chars[row] * NE + col];
  else          v = bigram_tab[(size_t)bigrams[row] * NE + (col - NE)];
  x[idx] = v;
}

// ---------------- 2: big input-projection GEMM with WMMA f32 ----------------
// G_dir[b*L+t][n] = sum_k x[src(b,t)][k] * W_ih[n][k] + bias[n]
// dir 0: src = (b,t); dir 1: src = (b, rev(b,t))  (packed-sequence reversal)
__global__ void __launch_bounds__(256)
ih_gemm_kernel(const float* __restrict__ x,
               const float* __restrict__ W_f, const float* __restrict__ b_f,
               const float* __restrict__ W_b, const float* __restrict__ b_b,
               const int* __restrict__ seq_len,
               float* __restrict__ G_f, float* __restrict__ G_b) {
  __shared__ float Wlds[NK * 16];                 // B-tile: [k][n-n0], 12.8KB
  int bid = blockIdx.x;                           // 2 * 128 * 25 = 6400 blocks
  int dir = bid / 3200;  int rem = bid - dir * 3200;
  int mg = rem / 25;     int nt = rem - mg * 25;
  int n0 = nt * 16;
  const float* W    = dir ? W_b : W_f;
  const float* bias = dir ? b_b : b_f;
  float*       G    = dir ? G_b : G_f;

  int tid = threadIdx.x;
  for (int i = tid; i < NK * 16; i += 256) {
    int k = i >> 4, nn = i & 15;
    Wlds[i] = W[(size_t)(n0 + nn) * NK + k];      // B[k][n] = W_ih[n][k]
  }
  __syncthreads();

  int wave = tid >> 5, lane = tid & 31;
  int lhi = lane >> 4, llo = lane & 15;
  int m0 = mg * 128 + wave * 16;                  // 8 waves x 16 rows

  // A-fragment source row for this lane (fixed across K loop)
  int m_a = m0 + llo;
  int src = m_a;
  if (dir) {
    int b = m_a >> 8, t = m_a & 255;
    int sl = seq_len[b];
    int rt = (t < sl) ? (sl - 1 - t) : t;
    src = (b << 8) + rt;
  }
  const float* arow = x + (size_t)src * NK;

  v8f acc = {};
  for (int k = 0; k < NK; k += 4) {
    int ka = k + 2 * lhi;                         // hi half-wave holds K+2
    v2f a;  a.x = arow[ka];              a.y = arow[ka + 1];
    v2f bf; bf.x = Wlds[ka * 16 + llo];  bf.y = Wlds[(ka + 1) * 16 + llo];
    acc = wmma4(a, bf, acc);
  }
#pragma unroll
  for (int r = 0; r < 8; ++r) {
    int m = m0 + r + 8 * lhi;
    int n = n0 + llo;
    G[(size_t)m * NG + n] = acc[r] + bias[n];
  }
}

// ---------------- 3: persistent recurrent LSTM scan (one WG per direction) ---
// 32 waves; waves 0..24 each own one fixed 16-col N-tile of the gate GEMM and
// keep their 25 W_hh B-fragments in registers for the whole scan. Gates live
// entirely in LDS (h + c + gates = 153.6KB of the 320KB WGP LDS).
__global__ void __launch_bounds__(1024)
lstm_kernel(const float* __restrict__ G_f, const float* __restrict__ G_b,
            const float* __restrict__ Whh_f, const float* __restrict__ Whh_b,
            float* __restrict__ hs_f, float* __restrict__ hs_b) {
  __shared__ float h_s[NB * NH];                  // 25.6 KB
  __shared__ float c_s[NB * NH];                  // 25.6 KB
  __shared__ float g_s[NB * NG];                  // 102.4 KB gates
  int dir = blockIdx.x;
  const float* G  = dir ? G_b   : G_f;
  const float* W  = dir ? Whh_b : Whh_f;
  float*       hs = dir ? hs_b  : hs_f;

  int tid = threadIdx.x;
  for (int i = tid; i < NB * NH; i += 1024) { h_s[i] = 0.f; c_s[i] = 0.f; }

  int wave = tid >> 5, lane = tid & 31;
  int lhi = lane >> 4, llo = lane & 15;
  int n0 = wave * 16;                             // this wave's N-tile (wave<25)

  // Preload this wave's W_hh B-fragments into registers (constant over t).
  v2f wfrag[25];
  if (wave < 25) {
#pragma unroll
    for (int kk = 0; kk < 25; ++kk) {
      int ka = 4 * kk + 2 * lhi;
      wfrag[kk].x = W[(size_t)(n0 + llo) * NH + ka];      // B[k][n]=Whh[n][k]
      wfrag[kk].y = W[(size_t)(n0 + llo) * NH + ka + 1];
    }
  }
  __syncthreads();

  for (int t = 0; t < NL; ++t) {
    // ---- g_s[64][400] = G[:,t,:] + h @ Whh^T  (WMMA f32 16x16x4) ----
    if (wave < 25) {
      for (int mt = 0; mt < 4; ++mt) {
        int m0 = mt * 16;
        v8f acc;
#pragma unroll
        for (int r = 0; r < 8; ++r) {
          int b = m0 + r + 8 * lhi;
          acc[r] = G[((size_t)(b << 8) + t) * NG + n0 + llo];
        }
#pragma unroll
        for (int kk = 0; kk < 25; ++kk) {
          int ka = 4 * kk + 2 * lhi;
          v2f a;  a.x = h_s[(m0 + llo) * NH + ka];
                  a.y = h_s[(m0 + llo) * NH + ka + 1];
          acc = wmma4(a, wfrag[kk], acc);
        }
#pragma unroll
        for (int r = 0; r < 8; ++r) {
          int m = m0 + r + 8 * lhi;
          g_s[m * NG + n0 + llo] = acc[r];
        }
      }
    }
    __syncthreads();
    // ---- elementwise cell update ----
    for (int e = tid; e < NB * NH; e += 1024) {
      int b = e / NH, j = e - b * NH;
      const float* gb = g_s + b * NG;
      float ig = gb[j], fg = gb[NH + j], gg = gb[2 * NH + j], og = gb[3 * NH + j];
      float si = 1.f / (1.f + __expf(-ig));
      float sf = 1.f / (1.f + __expf(-fg));
      float so = 1.f / (1.f + __expf(-og));
      float cn = sf * c_s[e] + si * tanhf(gg);
      float hn = so * tanhf(cn);
      c_s[e] = cn;  h_s[e] = hn;
      hs[((size_t)(b << 8) + t) * NH + j] = hn;
    }
    __syncthreads();                              // h ready before next step
  }
}

// ---------------- 4: FC (2H->33) + log_softmax ----------------
__global__ void __launch_bounds__(256)
fc_kernel(const float* __restrict__ hs_f, const float* __restrict__ hs_b,
          const int* __restrict__ seq_len,
          const float* __restrict__ fc_W, const float* __restrict__ fc_b,
          float* __restrict__ logits) {
  __shared__ float Wl[NC * NK];                   // 26.4 KB
  __shared__ float bl[NC];
  int tid = threadIdx.x;
  for (int i = tid; i < NC * NK; i += 256) Wl[i] = fc_W[i];
  if (tid < NC) bl[tid] = fc_b[tid];
  __syncthreads();

  int pos = blockIdx.x * 256 + tid;               // 0..16383
  int b = pos >> 8, t = pos & 255;
  int sl = seq_len[b];
  int rt = (t < sl) ? (sl - 1 - t) : t;           // un-reverse backward states
  const float* hf = hs_f + ((size_t)(b << 8) + t)  * NH;
  const float* hb = hs_b + ((size_t)(b << 8) + rt) * NH;

  float feat[NC];
#pragma unroll
  for (int c = 0; c < NC; ++c) feat[c] = bl[c];
  for (int k = 0; k < NH; ++k) {
    float v = hf[k];
#pragma unroll
    for (int c = 0; c < NC; ++c) feat[c] += v * Wl[c * NK + k];
  }
  for (int k = 0; k < NH; ++k) {
    float v = hb[k];
#pragma unroll
    for (int c = 0; c < NC; ++c) feat[c] += v * Wl[c * NK + NH + k];
  }
  float m = feat[0];
#pragma unroll
  for (int c = 1; c < NC; ++c) m = fmaxf(m, feat[c]);
  float s = 0.f;
#pragma unroll
  for (int c = 0; c < NC; ++c) s += __expf(feat[c] - m);
  float lse = m + __logf(s);
  float* out = logits + (size_t)pos * NC;
#pragma unroll
  for (int c = 0; c < NC; ++c) out[c] = feat[c] - lse;
}

// ---------------- 5: CRF NLL per batch element ----------------
__global__ void __launch_bounds__(64)
crf_kernel(const float* __restrict__ logits, const int* __restrict__ target,
           const int* __restrict__ seq_len, const float* __restrict__ trans,
           const float* __restrict__ start_s, const float* __restrict__ end_s,
           float* __restrict__ per_b) {
  __shared__ float tr[NC * NC];
  __shared__ float alpha[NC];
  __shared__ float tmp[NC];
  int b = blockIdx.x, tid = threadIdx.x;
  for (int i = tid; i < NC * NC; i += 64) tr[i] = trans[i];
  int sl = seq_len[b];
  const float* lg = logits + (size_t)(b << 8) * NC;   // [256][33]
  if (tid < NC) alpha[tid] = lg[tid] + start_s[tid];
  __syncthreads();

  for (int t = 1; t < NL; ++t) {
    float newv = 0.f, oldv = 0.f;
    if (tid < NC) {
      oldv = alpha[tid];
      float m = -1e30f;
      for (int i = 0; i < NC; ++i) m = fmaxf(m, alpha[i] + tr[i * NC + tid]);
      float s = 0.f;
      for (int i = 0; i < NC; ++i) s += __expf(alpha[i] + tr[i * NC + tid] - m);
      newv = m + __logf(s) + lg[t * NC + tid];
    }
    __syncthreads();
    if (tid < NC) alpha[tid] = (t < sl) ? newv : oldv;
    __syncthreads();
  }
  if (tid < NC) tmp[tid] = alpha[tid] + end_s[tid];
  __syncthreads();
  if (tid == 0) {
    float m = tmp[0];
    for (int i = 1; i < NC; ++i) m = fmaxf(m, tmp[i]);
    float s = 0.f;
    for (int i = 0; i < NC; ++i) s += __expf(tmp[i] - m);
    float Z = m + __logf(s);
    const int* tg = target + (b << 8);
    float gold = start_s[tg[0]] + end_s[tg[sl - 1]];
    for (int t = 0; t < sl; ++t) gold += lg[t * NC + tg[t]];
    for (int t = 1; t < sl; ++t) gold += tr[tg[t - 1] * NC + tg[t]];
    per_b[b] = Z - gold;
  }
}

// ---------------- 6: mean reduce ----------------
__global__ void reduce_kernel(const float* __restrict__ per_b, float* __restrict__ out) {
  if (threadIdx.x == 0 && blockIdx.x == 0) {
    float s = 0.f;
    for (int i = 0; i < NB; ++i) s += per_b[i];
    out[0] = s / (float)NB;
  }
}

extern "C" void kernel_launch(void* const* d_in, const int* in_sizes, int n_in,
                              void* d_out, int out_size, void* d_ws, size_t ws_size,
                              hipStream_t stream) {
  const int*   chars      = (const int*)d_in[0];
  const int*   bigrams    = (const int*)d_in[1];
  const int*   seq_len    = (const int*)d_in[2];
  const int*   target     = (const int*)d_in[3];
  const float* char_tab   = (const float*)d_in[4];
  const float* bigram_tab = (const float*)d_in[5];
  const float* W_ih_f     = (const float*)d_in[6];
  const float* W_hh_f     = (const float*)d_in[7];
  const float* b_f        = (const float*)d_in[8];
  const float* W_ih_b     = (const float*)d_in[9];
  const float* W_hh_b     = (const float*)d_in[10];
  const float* b_b        = (const float*)d_in[11];
  const float* fc_W       = (const float*)d_in[12];
  const float* fc_b       = (const float*)d_in[13];
  const float* trans_m    = (const float*)d_in[14];
  const float* start_s    = (const float*)d_in[15];
  const float* end_s      = (const float*)d_in[16];

  float* ws     = (float*)d_ws;
  float* x      = ws + X_OFF;
  float* G_f    = ws + GF_OFF;
  float* G_b    = ws + GB_OFF;
  float* hs_f   = ws + HF_OFF;
  float* hs_b   = ws + HB_OFF;
  float* logits = ws + LOGITS_OFF;
  float* perb   = ws + PERB_OFF;
  float* out    = (float*)d_out;

  embed_kernel<<<(16384 * NK) / 256, 256, 0, stream>>>(chars, bigrams, char_tab,
                                                       bigram_tab, x);
  ih_gemm_kernel<<<6400, 256, 0, stream>>>(x, W_ih_f, b_f, W_ih_b, b_b, seq_len,
                                           G_f, G_b);
  lstm_kernel<<<2, 1024, 0, stream>>>(G_f, G_b, W_hh_f, W_hh_b, hs_f, hs_b);
  fc_kernel<<<64, 256, 0, stream>>>(hs_f, hs_b, seq_len, fc_W, fc_b, logits);
  crf_kernel<<<64, 64, 0, stream>>>(logits, target, seq_len, trans_m, start_s,
                                    end_s, perb);
  reduce_kernel<<<1, 32, 0, stream>>>(perb, out);
}